// FTDGNN_10256381903670
// MI455X (gfx1250) — compile-verified
//
#include <hip/hip_runtime.h>
#include <math.h>

#define B_    4
#define ROI_  90
#define T_    200
#define H_    3
#define F_    96
#define HID_  96
#define G3_   288      // 3*HID
#define BH_   12       // B_*H_
#define NP_   96       // padded roi/feature tile dim
#define STR_  100      // LDS row stride in floats (bank-conflict friendly)
#define GSTR_ 292      // gate buffer row stride
#define TPB_  288      // 9 wave32s

typedef float v2f __attribute__((ext_vector_type(2)));
typedef float v8f __attribute__((ext_vector_type(8)));

__device__ __forceinline__ v8f wmma4(v2f a, v2f b, v8f c) {
  return __builtin_amdgcn_wmma_f32_16x16x4_f32(false, a, false, b, (short)0, c,
                                               false, false);
}

// 2x2 super-tile (32x32 output), BOTH operands as contiguous v2f reads:
// A row-major [m][k] stride STR_; B transposed Bt[n][k] stride STR_.
// Per k-step: 4 v2f loads feed 4 independent WMMAs.
__device__ __forceinline__ void mm2x2_trB(const float* As, const float* Bt,
                                          int sr, int sc, int lane, v8f acc[4]) {
  const int half = lane >> 4;
  const int l15  = lane & 15;
  const float* a0 = As + (sr * 32 + l15) * STR_ + (half << 1);
  const float* a1 = a0 + 16 * STR_;
  const float* b0 = Bt + (sc * 32 + l15) * STR_ + (half << 1);
  const float* b1 = b0 + 16 * STR_;
  #pragma unroll
  for (int j = 0; j < 4; ++j) acc[j] = (v8f){0.f,0.f,0.f,0.f,0.f,0.f,0.f,0.f};
  #pragma unroll
  for (int k0 = 0; k0 < NP_; k0 += 4) {
    v2f av0 = *(const v2f*)(a0 + k0);
    v2f av1 = *(const v2f*)(a1 + k0);
    v2f bv0 = *(const v2f*)(b0 + k0);
    v2f bv1 = *(const v2f*)(b1 + k0);
    acc[0] = wmma4(av0, bv0, acc[0]);
    acc[1] = wmma4(av0, bv1, acc[1]);
    acc[2] = wmma4(av1, bv0, acc[2]);
    acc[3] = wmma4(av1, bv1, acc[3]);
  }
}

// 1x4 row of tiles (16 x 64 output), shared A row, B transposed (Bt[n][k]).
__device__ __forceinline__ void mm1x4_trB(const float* As, const float* Bt,
                                          int ntBase, int lane, v8f acc[4]) {
  const int half = lane >> 4;
  const int l15  = lane & 15;
  const float* aRow = As + l15 * STR_ + (half << 1);
  const float* bRow = Bt + (ntBase * 16 + l15) * STR_ + (half << 1);
  #pragma unroll
  for (int j = 0; j < 4; ++j) acc[j] = (v8f){0.f,0.f,0.f,0.f,0.f,0.f,0.f,0.f};
  #pragma unroll
  for (int k0 = 0; k0 < F_; k0 += 4) {
    v2f av = *(const v2f*)(aRow + k0);
    #pragma unroll
    for (int j = 0; j < 4; ++j) {
      v2f bv = *(const v2f*)(bRow + j * 16 * STR_ + k0);
      acc[j] = wmma4(av, bv, acc[j]);
    }
  }
}

__device__ __forceinline__ void mm1x2_trB(const float* As, const float* Bt,
                                          int ntBase, int lane, v8f acc[2]) {
  const int half = lane >> 4;
  const int l15  = lane & 15;
  const float* aRow = As + l15 * STR_ + (half << 1);
  const float* bRow = Bt + (ntBase * 16 + l15) * STR_ + (half << 1);
  #pragma unroll
  for (int j = 0; j < 2; ++j) acc[j] = (v8f){0.f,0.f,0.f,0.f,0.f,0.f,0.f,0.f};
  #pragma unroll
  for (int k0 = 0; k0 < F_; k0 += 4) {
    v2f av = *(const v2f*)(aRow + k0);
    #pragma unroll
    for (int j = 0; j < 2; ++j) {
      v2f bv = *(const v2f*)(bRow + j * 16 * STR_ + k0);
      acc[j] = wmma4(av, bv, acc[j]);
    }
  }
}

// ---------------- Kernel A: masked aggregation + 2-layer MLP (fused) ----------------
// One block per (b,t,h). Wave w owns the 2x2 super-tile (sr, sc) = (w/3, w%3).
// LDS: xs (x^T tile, later h1 row-major) | ms (mask -> agg) | ws (W1^T -> W2^T).
__global__ void __launch_bounds__(TPB_)
ftdgnn_agg_mlp(const float* __restrict__ x, const float* __restrict__ a,
               const float* __restrict__ eps,
               const float* __restrict__ W1, const float* __restrict__ b1,
               const float* __restrict__ g1, const float* __restrict__ be1,
               const float* __restrict__ W2, const float* __restrict__ b2,
               const float* __restrict__ g2, const float* __restrict__ be2,
               float* __restrict__ hbuf) {
  extern __shared__ float smem[];
  float* xs = smem;                  // stage1 B: x^T [f][roi]; stage3 A: h1 [m][k]
  float* ms = xs + NP_ * STR_;       // stage1 A: mask [i][j]; then agg [m][k]
  float* ws = ms + NP_ * STR_;       // W1^T then W2^T  [n][k]

  const int tid  = threadIdx.x;
  const int lane = tid & 31;
  const int wave = tid >> 5;         // 0..8
  const int half = lane >> 4;
  const int l15  = lane & 15;
  const int sr   = wave / 3;         // super-tile row (0..2)
  const int sc   = wave % 3;         // super-tile col (0..2)

  const int bth = blockIdx.x;
  const int b   = bth / (T_ * H_);
  const int th  = bth % (T_ * H_);
  const int t   = th / H_;
  const int hh  = th % H_;

  for (int idx = tid; idx < NP_ * NP_; idx += TPB_) {   // 9216/288 = 32 exact
    const int r = idx / NP_, c = idx % NP_;
    float xv = 0.f;
    if (r < ROI_)
      xv = x[(((size_t)(b * ROI_ + r) * T_ + t) * H_ + hh) * F_ + c];
    xs[c * STR_ + r] = xv;                              // transposed: xs[f][roi]
    float mv = 0.f;
    if (r < ROI_ && c < ROI_) {
      const float av = a[(((size_t)(b * ROI_ + r) * T_ + t) * H_ + hh) * ROI_ + c];
      mv = (av != 0.f) ? 1.f : 0.f;
    }
    ms[r * STR_ + c] = mv;
    ws[c * STR_ + r] = W1[r * HID_ + c];                // transposed: ws[n][k]
  }
  __syncthreads();

  const float epsv = eps[0];
  const float rs   = rsqrtf(1.0f + 1e-5f);   // BN eval: mean 0, var 1

  v8f acc[4];

  // stage 1: agg = mask @ x + eps * x   (B = x supplied transposed)
  mm2x2_trB(ms, xs, sr, sc, lane, acc);
  __syncthreads();                           // all reads of ms done
  #pragma unroll
  for (int di = 0; di < 2; ++di)
    #pragma unroll
    for (int dj = 0; dj < 2; ++dj) {
      const int n = (2 * sc + dj) * 16 + l15;
      #pragma unroll
      for (int v = 0; v < 8; ++v) {
        const int m = (2 * sr + di) * 16 + v + 8 * half;
        ms[m * STR_ + n] = acc[di * 2 + dj][v] + epsv * xs[n * STR_ + m];
      }
    }
  __syncthreads();

  // stage 2: h1 = elu(bn1(agg @ W1 + b1))   (W1 in LDS transposed)
  mm2x2_trB(ms, ws, sr, sc, lane, acc);
  __syncthreads();                           // reads of ws(W1^T)/xs done
  #pragma unroll
  for (int di = 0; di < 2; ++di)
    #pragma unroll
    for (int dj = 0; dj < 2; ++dj) {
      const int n = (2 * sc + dj) * 16 + l15;
      const float sc1 = g1[n] * rs, sh1 = be1[n], bb1 = b1[n];
      #pragma unroll
      for (int v = 0; v < 8; ++v) {
        const int m = (2 * sr + di) * 16 + v + 8 * half;
        float z = (acc[di * 2 + dj][v] + bb1) * sc1 + sh1;
        z = (z > 0.f) ? z : expm1f(z);                // ELU(alpha=1)
        xs[m * STR_ + n] = z;                         // h1 row-major -> xs
      }
    }
  for (int idx = tid; idx < NP_ * NP_; idx += TPB_) {
    const int r = idx / NP_, c = idx % NP_;
    ws[c * STR_ + r] = W2[r * F_ + c];                // W2^T -> ws
  }
  __syncthreads();

  // stage 3: h = relu(bn2(h1 @ W2 + b2))   [relu(elu(z)) == relu(z)]
  mm2x2_trB(xs, ws, sr, sc, lane, acc);
  #pragma unroll
  for (int di = 0; di < 2; ++di)
    #pragma unroll
    for (int dj = 0; dj < 2; ++dj) {
      const int n = (2 * sc + dj) * 16 + l15;
      const float sc2 = g2[n] * rs, sh2 = be2[n], bb2 = b2[n];
      #pragma unroll
      for (int v = 0; v < 8; ++v) {
        const int m = (2 * sr + di) * 16 + v + 8 * half;
        if (m < ROI_) {
          float z = (acc[di * 2 + dj][v] + bb2) * sc2 + sh2;
          z = (z > 0.f) ? z : 0.f;
          // layout for GRU: hbuf[roi][t][bh][f]
          hbuf[((size_t)(m * T_ + t) * BH_ + (b * H_ + hh)) * F_ + n] = z;
        }
      }
    }
}

// ---------------- Kernel B: per-ROI GRU over T steps ----------------
// One block per ROI. Wih/Whh resident in LDS (2 x 115 KB) for all 200 steps.
__global__ void __launch_bounds__(TPB_)
ftdgnn_gru(const float* __restrict__ Wih, const float* __restrict__ Whh,
           const float* __restrict__ bih, const float* __restrict__ bhh,
           const float* __restrict__ hbuf, float* __restrict__ out) {
  extern __shared__ float smem[];
  float* WihS = smem;                      // G3_ x STR_
  float* WhhS = WihS + G3_ * STR_;         // G3_ x STR_
  float* hS   = WhhS + G3_ * STR_;         // 16 x STR_ (rows 12..15 stay 0)
  float* xS   = hS   + 16 * STR_;          // 16 x STR_
  float* gx   = xS   + 16 * STR_;          // 16 x GSTR_ : x_t @ Wih^T
  float* gh   = gx   + 16 * GSTR_;         // 16 x GSTR_ : h   @ Whh^T
  float* bihS = gh   + 16 * GSTR_;         // G3_
  float* bhhS = bihS + G3_;                // G3_

  const int tid  = threadIdx.x;
  const int lane = tid & 31;
  const int wave = tid >> 5;               // 0..8
  const int half = lane >> 4;
  const int l15  = lane & 15;
  const int r    = blockIdx.x;             // ROI index

  for (int idx = tid; idx < G3_ * F_; idx += TPB_) {    // 27648/288 = 96 exact
    const int g = idx / F_, k = idx % F_;
    WihS[g * STR_ + k] = Wih[((size_t)r * G3_ + g) * F_ + k];
    WhhS[g * STR_ + k] = Whh[((size_t)r * G3_ + g) * HID_ + k];
  }
  for (int idx = tid; idx < G3_; idx += TPB_) {
    bihS[idx] = bih[r * G3_ + idx];
    bhhS[idx] = bhh[r * G3_ + idx];
  }
  for (int idx = tid; idx < 16 * STR_; idx += TPB_) { hS[idx] = 0.f; xS[idx] = 0.f; }
  __syncthreads();
  for (int idx = tid; idx < BH_ * F_; idx += TPB_) {    // 1152/288 = 4 exact
    const int m = idx / F_, k = idx % F_;
    xS[m * STR_ + k] = hbuf[((size_t)(r * T_) * BH_ + m) * F_ + k];
  }
  __syncthreads();

  for (int t = 0; t < T_; ++t) {
    // 36 gate tiles; wave-uniform paths so the A source is static per path:
    //   waves 0..3 : xp tiles 4w..4w+3        (gx = xS @ WihS^T)
    //   wave  4    : xp tiles 16,17 + hp 0,1
    //   waves 5..8 : hp tiles 4w-18..4w-15    (gh = hS @ WhhS^T)
    if (wave < 4) {
      v8f acc[4];
      mm1x4_trB(xS, WihS, wave * 4, lane, acc);
      #pragma unroll
      for (int j = 0; j < 4; ++j) {
        const int n = (wave * 4 + j) * 16 + l15;
        #pragma unroll
        for (int v = 0; v < 8; ++v)
          gx[(v + 8 * half) * GSTR_ + n] = acc[j][v];
      }
    } else if (wave > 4) {
      v8f acc[4];
      mm1x4_trB(hS, WhhS, wave * 4 - 18, lane, acc);
      #pragma unroll
      for (int j = 0; j < 4; ++j) {
        const int n = (wave * 4 - 18 + j) * 16 + l15;
        #pragma unroll
        for (int v = 0; v < 8; ++v)
          gh[(v + 8 * half) * GSTR_ + n] = acc[j][v];
      }
    } else {
      v8f ax[2], ah[2];
      mm1x2_trB(xS, WihS, 16, lane, ax);
      mm1x2_trB(hS, WhhS, 0, lane, ah);
      #pragma unroll
      for (int j = 0; j < 2; ++j) {
        const int nx = (16 + j) * 16 + l15;
        const int nh = j * 16 + l15;
        #pragma unroll
        for (int v = 0; v < 8; ++v) {
          gx[(v + 8 * half) * GSTR_ + nx] = ax[j][v];
          gh[(v + 8 * half) * GSTR_ + nh] = ah[j][v];
        }
      }
    }
    __syncthreads();

    // gates, state update, output, next-x prefetch
    for (int idx = tid; idx < BH_ * HID_; idx += TPB_) {
      const int m = idx / HID_, f = idx % HID_;
      const float xr = gx[m * GSTR_ + f]             + bihS[f];
      const float xz = gx[m * GSTR_ + HID_ + f]      + bihS[HID_ + f];
      const float xn = gx[m * GSTR_ + 2 * HID_ + f]  + bihS[2 * HID_ + f];
      const float hr = gh[m * GSTR_ + f]             + bhhS[f];
      const float hz = gh[m * GSTR_ + HID_ + f]      + bhhS[HID_ + f];
      const float hn = gh[m * GSTR_ + 2 * HID_ + f]  + bhhS[2 * HID_ + f];
      const float rg = 1.f / (1.f + expf(-(xr + hr)));
      const float zg = 1.f / (1.f + expf(-(xz + hz)));
      const float ng = tanhf(xn + rg * hn);
      const float h2 = (1.f - zg) * ng + zg * hS[m * STR_ + f];
      hS[m * STR_ + f] = h2;
      const int bb = m / H_, hh = m % H_;
      out[(((size_t)(bb * ROI_ + r) * T_ + t) * H_ + hh) * HID_ + f] = h2;
    }
    if (t + 1 < T_) {
      for (int idx = tid; idx < BH_ * F_; idx += TPB_) {
        const int m = idx / F_, k = idx % F_;
        xS[m * STR_ + k] = hbuf[((size_t)(r * T_ + (t + 1)) * BH_ + m) * F_ + k];
      }
      if (t + 2 < T_) {
        // warm GL2/WGP$ for the step after next (global_prefetch_b8)
        const int pidx = tid & (BH_ - 1);
        __builtin_prefetch(&hbuf[((size_t)(r * T_ + (t + 2)) * BH_ + pidx) * F_], 0, 1);
      }
    }
    __syncthreads();
  }
}

extern "C" void kernel_launch(void* const* d_in, const int* in_sizes, int n_in,
                              void* d_out, int out_size, void* d_ws, size_t ws_size,
                              hipStream_t stream) {
  const float* x   = (const float*)d_in[0];
  const float* a   = (const float*)d_in[1];
  const float* eps = (const float*)d_in[2];
  const float* W1  = (const float*)d_in[3];
  const float* b1  = (const float*)d_in[4];
  const float* g1  = (const float*)d_in[5];
  const float* be1 = (const float*)d_in[6];
  const float* W2  = (const float*)d_in[7];
  const float* b2  = (const float*)d_in[8];
  const float* g2  = (const float*)d_in[9];
  const float* be2 = (const float*)d_in[10];
  const float* Wih = (const float*)d_in[11];
  const float* Whh = (const float*)d_in[12];
  const float* bih = (const float*)d_in[13];
  const float* bhh = (const float*)d_in[14];
  float* out  = (float*)d_out;
  float* hbuf = (float*)d_ws;   // needs ROI*T*BH*F floats (~83 MB)

  const size_t ldsA = (size_t)3 * NP_ * STR_ * sizeof(float);          // 115.2 KB
  ftdgnn_agg_mlp<<<B_ * T_ * H_, TPB_, ldsA, stream>>>(
      x, a, eps, W1, b1, g1, be1, W2, b2, g2, be2, hbuf);

  const size_t ldsB = (size_t)(2 * G3_ * STR_ + 2 * 16 * STR_ +
                               2 * 16 * GSTR_ + 2 * G3_) * sizeof(float); // ~283 KB
  ftdgnn_gru<<<ROI_, TPB_, ldsB, stream>>>(Wih, Whh, bih, bhh, hbuf, out);
}